// GCN_309237645923
// MI455X (gfx1250) — compile-verified
//
#include <hip/hip_runtime.h>

typedef float v2f __attribute__((ext_vector_type(2)));
typedef float v8f __attribute__((ext_vector_type(8)));

#define F_IN 512
#define HID  16
#define C_OUT 40
#define NPAD 48

// ---------------- degree / norm ----------------
__global__ void k_init_deg(float* deg, int n) {
    int i = blockIdx.x * blockDim.x + threadIdx.x;
    if (i < n) deg[i] = 1.0f;            // self-loop
}

__global__ void k_deg_edges(float* deg, const int* __restrict__ dst, int e) {
    int i = blockIdx.x * blockDim.x + threadIdx.x;
    if (i < e) unsafeAtomicAdd(&deg[dst[i]], 1.0f);
}

__global__ void k_finalize_dinv(float* deg, int n) {
    int i = blockIdx.x * blockDim.x + threadIdx.x;
    if (i < n) deg[i] = rsqrtf(deg[i]);  // deg >= 1 always (self-loop)
}

// ---------------- GEMM1: h = x @ W1  (100000x512 @ 512x16), fp32 WMMA ----------------
__global__ __launch_bounds__(256) void k_gemm1_wmma(const float* __restrict__ x,
                                                    const float* __restrict__ W1,
                                                    float* __restrict__ h, int n_rows) {
    __shared__ float lds_w[F_IN * HID];                  // 32 KB
    for (int i = threadIdx.x; i < F_IN * HID; i += 256) lds_w[i] = W1[i];
    __syncthreads();

    const int wave = threadIdx.x >> 5;
    const int lane = threadIdx.x & 31;
    const int tile = blockIdx.x * 8 + wave;
    if (tile * 16 >= n_rows) return;                     // whole-wave guard: EXEC stays all-1

    const int row  = tile * 16 + (lane & 15);
    const int kb   = (lane >> 4) << 1;                   // 0 or 2
    const int n    = lane & 15;
    const int half = lane >> 4;

    v8f acc = {};
    const float* xrow = x + (long)row * F_IN + kb;
    const float* wcol = lds_w + kb * HID + n;

    #pragma unroll 8
    for (int k = 0; k < F_IN; k += 4) {
        v2f a = *(const v2f*)(xrow + k);                 // {K=kb, K=kb+1}
        v2f b;
        b.x = wcol[k * HID];                             // W1[(k+kb)  ][n]
        b.y = wcol[(k + 1) * HID];                       // W1[(k+kb+1)][n]
        acc = __builtin_amdgcn_wmma_f32_16x16x4_f32(false, a, false, b,
                                                    (short)0, acc, false, false);
    }

    #pragma unroll
    for (int i = 0; i < 8; ++i)
        h[(long)(tile * 16 + i + 8 * half) * HID + n] = acc[i];
}

// ---------------- layer-1 aggregation ----------------
__global__ void k_init_agg16(const float* __restrict__ h, const float* __restrict__ dinv,
                             const float* __restrict__ b1, float* __restrict__ hagg, int n) {
    int i = blockIdx.x * blockDim.x + threadIdx.x;
    if (i < n * HID) {
        int node = i >> 4, f = i & 15;
        float dv = dinv[node];
        hagg[i] = h[i] * dv * dv + b1[f];                // self-loop + bias (bias is additive-safe)
    }
}

__global__ void k_edge_agg16(const float* __restrict__ h, const float* __restrict__ dinv,
                             const int* __restrict__ src, const int* __restrict__ dst,
                             float* __restrict__ hagg, int e) {
    long t = (long)blockIdx.x * blockDim.x + threadIdx.x;
    long ee = t >> 4;
    int  f  = (int)(t & 15);
    if (ee < e) {
        int s = src[ee], d = dst[ee];
        float norm = dinv[s] * dinv[d];
        unsafeAtomicAdd(&hagg[(long)d * HID + f], h[(long)s * HID + f] * norm);
    }
}

__global__ void k_relu(float* p, int n) {
    int i = blockIdx.x * blockDim.x + threadIdx.x;
    if (i < n) p[i] = fmaxf(p[i], 0.0f);
}

// ---------------- GEMM2: g = h2 @ W2  (100000x16 @ 16x40 padded to 48), fp32 WMMA ----------------
__global__ __launch_bounds__(256) void k_gemm2_wmma(const float* __restrict__ h2,
                                                    const float* __restrict__ W2,
                                                    float* __restrict__ g, int n_rows) {
    __shared__ float lds_w[HID * NPAD];                  // 3 KB, zero-padded 40->48
    for (int i = threadIdx.x; i < HID * NPAD; i += 256) {
        int kk = i / NPAD, nn = i % NPAD;
        lds_w[i] = (nn < C_OUT) ? W2[kk * C_OUT + nn] : 0.0f;
    }
    __syncthreads();

    const int wave = threadIdx.x >> 5;
    const int lane = threadIdx.x & 31;
    const int tile = blockIdx.x * 8 + wave;
    if (tile * 16 >= n_rows) return;

    const int row  = tile * 16 + (lane & 15);
    const int kb   = (lane >> 4) << 1;
    const int n    = lane & 15;
    const int half = lane >> 4;

    v8f acc0 = {}, acc1 = {}, acc2 = {};
    const float* arow = h2 + (long)row * HID + kb;

    #pragma unroll
    for (int k = 0; k < HID; k += 4) {
        v2f a = *(const v2f*)(arow + k);
        const float* w0 = lds_w + (k + kb) * NPAD + n;
        const float* w1 = w0 + NPAD;
        v2f b0, b1, b2;
        b0.x = w0[0];  b0.y = w1[0];
        b1.x = w0[16]; b1.y = w1[16];
        b2.x = w0[32]; b2.y = w1[32];
        acc0 = __builtin_amdgcn_wmma_f32_16x16x4_f32(false, a, false, b0, (short)0, acc0, false, false);
        acc1 = __builtin_amdgcn_wmma_f32_16x16x4_f32(false, a, false, b1, (short)0, acc1, false, false);
        acc2 = __builtin_amdgcn_wmma_f32_16x16x4_f32(false, a, false, b2, (short)0, acc2, false, false);
    }

    #pragma unroll
    for (int i = 0; i < 8; ++i) {
        int r = tile * 16 + i + 8 * half;
        g[(long)r * C_OUT + n]      = acc0[i];
        g[(long)r * C_OUT + 16 + n] = acc1[i];
        if (n < 8) g[(long)r * C_OUT + 32 + n] = acc2[i];
    }
}

// ---------------- layer-2 aggregation (writes d_out) ----------------
__global__ void k_init_agg40(const float* __restrict__ g, const float* __restrict__ dinv,
                             const float* __restrict__ b2, float* __restrict__ out, int n) {
    int i = blockIdx.x * blockDim.x + threadIdx.x;
    if (i < n * C_OUT) {
        int node = i / C_OUT, f = i % C_OUT;
        float dv = dinv[node];
        out[i] = g[i] * dv * dv + b2[f];                 // self-loop + final bias
    }
}

__global__ void k_edge_agg40(const float* __restrict__ g, const float* __restrict__ dinv,
                             const int* __restrict__ src, const int* __restrict__ dst,
                             float* __restrict__ out, int e) {
    long t = (long)blockIdx.x * blockDim.x + threadIdx.x;
    long ee = t / C_OUT;
    int  f  = (int)(t % C_OUT);
    if (ee < e) {
        int s = src[ee], d = dst[ee];
        float norm = dinv[s] * dinv[d];
        unsafeAtomicAdd(&out[(long)d * C_OUT + f], g[(long)s * C_OUT + f] * norm);
    }
}

extern "C" void kernel_launch(void* const* d_in, const int* in_sizes, int n_in,
                              void* d_out, int out_size, void* d_ws, size_t ws_size,
                              hipStream_t stream) {
    const float* x  = (const float*)d_in[0];
    const int*   ei = (const int*)d_in[1];
    const float* W1 = (const float*)d_in[2];
    const float* b1 = (const float*)d_in[3];
    const float* W2 = (const float*)d_in[4];
    const float* b2 = (const float*)d_in[5];
    float* out = (float*)d_out;

    const int N = in_sizes[0] / F_IN;        // 100000
    const int E = in_sizes[1] / 2;           // 3200000
    const int* src = ei;                     // edge_index[0]
    const int* dst = ei + E;                 // edge_index[1]

    float* ws   = (float*)d_ws;
    float* dinv = ws;                                    // N
    float* h    = ws + N;                                // N*16
    float* hagg = ws + N + (size_t)N * HID;              // N*16
    float* g    = ws + N + (size_t)2 * N * HID;          // N*40

    const int T = 256;
    const int tiles = (N + 15) / 16;

    k_init_deg      <<<(N + T - 1) / T, T, 0, stream>>>(dinv, N);
    k_deg_edges     <<<(E + T - 1) / T, T, 0, stream>>>(dinv, dst, E);
    k_finalize_dinv <<<(N + T - 1) / T, T, 0, stream>>>(dinv, N);

    k_gemm1_wmma    <<<(tiles + 7) / 8, T, 0, stream>>>(x, W1, h, N);

    k_init_agg16    <<<((long)N * HID + T - 1) / T, T, 0, stream>>>(h, dinv, b1, hagg, N);
    k_edge_agg16    <<<((long)E * HID + T - 1) / T, T, 0, stream>>>(h, dinv, src, dst, hagg, E);
    k_relu          <<<((long)N * HID + T - 1) / T, T, 0, stream>>>(hagg, N * HID);

    k_gemm2_wmma    <<<(tiles + 7) / 8, T, 0, stream>>>(hagg, W2, g, N);

    k_init_agg40    <<<((long)N * C_OUT + T - 1) / T, T, 0, stream>>>(g, dinv, b2, out, N);
    k_edge_agg40    <<<((long)E * C_OUT + T - 1) / T, T, 0, stream>>>(g, dinv, src, dst, out, E);
}